// InvolutionLayer_8796093022818
// MI455X (gfx1250) — compile-verified
//
#include <hip/hip_runtime.h>

typedef __attribute__((ext_vector_type(2))) float v2f;
typedef __attribute__((ext_vector_type(4))) float v4f;
typedef __attribute__((ext_vector_type(8))) float v8f;

#define Bn   4
#define Hn   128
#define Wn   128
#define Cn   256
#define Gn   16
#define K2n  9
#define Crn  64
#define KGn  (K2n * Gn)          // 144
#define NPIX (Bn * Hn * Wn)      // 65536
#define BN_EPS 1e-3f

// ---------------------------------------------------------------------------
// Kernel 1: fused  h = relu(BN(x@w1 + b1)) ;  kern = h@w2 + b2
// One wave handles a tile of 16 pixels (M=16). fp32 WMMA 16x16x4.
//   A (16x4):  lane m=lane&15 holds row M=m; K-halves split by lane>>4.
//   B (4x16):  lanes 0-15 hold K=0 (V0)/K=1 (V1); lanes 16-31 K=2/K=3.
//   C/D (16x16): lane n=lane&15, VGPR r -> M = r + 8*(lane>>4).
// ---------------------------------------------------------------------------
__global__ __launch_bounds__(256) void invol_kergen(
    const float* __restrict__ x,  const float* __restrict__ w1,
    const float* __restrict__ b1, const float* __restrict__ gamma,
    const float* __restrict__ beta, const float* __restrict__ mean,
    const float* __restrict__ var,  const float* __restrict__ w2,
    const float* __restrict__ b2,   float* __restrict__ kout)
{
    __shared__ float h_lds[8][16 * Crn];            // 32 KB: per-wave h tile

    const int lane = threadIdx.x & 31;
    const int wave = threadIdx.x >> 5;
    const int tile = blockIdx.x * 8 + wave;         // 4096 tiles total
    const int p0   = tile * 16;                     // first pixel of tile
    const int m    = lane & 15;                     // M row / N col index
    const int hi   = lane >> 4;                     // half-wave select
    const int dk   = hi * 2;                        // K offset for this half

    // ---- Phase A: h[16 x 64] = x[16 x 256] @ w1[256 x 64], + bias/BN/ReLU
    const float* xrow = x + (size_t)(p0 + m) * Cn + dk;
    for (int nt = 0; nt < 4; ++nt) {                // 4 N-tiles of 16 cols
        const int n0 = nt * 16;
        v8f acc = {};
        for (int kk = 0; kk < Cn / 4; ++kk) {       // 64 K-steps of 4
            const int kb = kk * 4;
            v2f a = *(const v2f*)(xrow + kb);       // x channels kb+dk, kb+dk+1
            v2f bf;
            bf.x = w1[(kb + dk    ) * Crn + n0 + m];
            bf.y = w1[(kb + dk + 1) * Crn + n0 + m];
            acc = __builtin_amdgcn_wmma_f32_16x16x4_f32(
                      false, a, false, bf, (short)0, acc, false, false);
        }
        const int n = n0 + m;
        const float sc = gamma[n] * rsqrtf(var[n] + BN_EPS);
        const float sh = beta[n] - mean[n] * sc;
        const float bb = b1[n];
#pragma unroll
        for (int r = 0; r < 8; ++r) {
            float v = (acc[r] + bb) * sc + sh;      // BN(x@w1 + b1)
            v = v > 0.0f ? v : 0.0f;                // ReLU
            h_lds[wave][(r + hi * 8) * Crn + n] = v;
        }
    }
    __syncthreads();

    // ---- Phase B: kern[16 x 144] = h[16 x 64] @ w2[64 x 144] + b2
    for (int nt = 0; nt < 9; ++nt) {                // 9 N-tiles of 16 cols
        const int n2 = nt * 16 + m;
        v8f acc = {};
        for (int kk = 0; kk < Crn / 4; ++kk) {      // 16 K-steps of 4
            const int kb = kk * 4;
            v2f a = *(const v2f*)(&h_lds[wave][m * Crn + kb + dk]);
            v2f bf;
            bf.x = w2[(kb + dk    ) * KGn + n2];
            bf.y = w2[(kb + dk + 1) * KGn + n2];
            acc = __builtin_amdgcn_wmma_f32_16x16x4_f32(
                      false, a, false, bf, (short)0, acc, false, false);
        }
        const float bb = b2[n2];
#pragma unroll
        for (int r = 0; r < 8; ++r)
            kout[(size_t)(p0 + r + hi * 8) * KGn + n2] = acc[r] + bb;
    }
}

// ---------------------------------------------------------------------------
// Kernel 2: involution gather.  out[p,c] = sum_k x[nbr_k(p), c] * kern[p, k*16 + c%16]
// One thread = 4 contiguous channels of one pixel (all float4, 16B aligned).
// ---------------------------------------------------------------------------
__global__ __launch_bounds__(256) void invol_apply(
    const float* __restrict__ x, const float* __restrict__ kern,
    float* __restrict__ out)
{
    const int t  = blockIdx.x * 256 + threadIdx.x;  // pixel*64 + cgroup
    const int ci = t & 63;
    const int p  = t >> 6;
    const int c0 = ci * 4;
    const int g0 = c0 & 15;

    const int b   = p >> 14;                        // / (128*128)
    const int rem = p & 16383;
    const int y   = rem >> 7;
    const int xx  = rem & 127;

    const float* kp = kern + (size_t)p * KGn + g0;
    v4f acc = {};
#pragma unroll
    for (int ki = 0; ki < 3; ++ki) {
        const int yy = y + ki - 1;
        if (yy < 0 || yy >= Hn) continue;           // zero padding
#pragma unroll
        for (int kj = 0; kj < 3; ++kj) {
            const int xj = xx + kj - 1;
            if (xj < 0 || xj >= Wn) continue;       // zero padding
            const v4f xv = *(const v4f*)(x + ((size_t)((b * Hn + yy) * Wn + xj)) * Cn + c0);
            const v4f kv = *(const v4f*)(kp + (ki * 3 + kj) * Gn);
            acc += xv * kv;
        }
    }
    *(v4f*)(out + (size_t)p * Cn + c0) = acc;
}

// ---------------------------------------------------------------------------
extern "C" void kernel_launch(void* const* d_in, const int* in_sizes, int n_in,
                              void* d_out, int out_size, void* d_ws, size_t ws_size,
                              hipStream_t stream)
{
    const float* x     = (const float*)d_in[0];
    const float* w1    = (const float*)d_in[1];
    const float* b1    = (const float*)d_in[2];
    const float* gamma = (const float*)d_in[3];
    const float* beta  = (const float*)d_in[4];
    const float* mean  = (const float*)d_in[5];
    const float* var   = (const float*)d_in[6];
    const float* w2    = (const float*)d_in[7];
    const float* b2    = (const float*)d_in[8];

    float* out  = (float*)d_out;                    // (B,H,W,C)    = 16777216 floats
    float* kout = out + (size_t)NPIX * Cn;          // (B,H,W,9,1,16) = 9437184 floats

    // 4096 pixel-tiles of 16, 8 waves (tiles) per 256-thread block -> 512 blocks
    invol_kergen<<<NPIX / 16 / 8, 256, 0, stream>>>(
        x, w1, b1, gamma, beta, mean, var, w2, b2, kout);

    // 65536 pixels * 64 channel-groups -> 16384 blocks of 256
    invol_apply<<<(NPIX * 64) / 256, 256, 0, stream>>>(x, kout, out);
}